// _TopKRouter_25872882991285
// MI455X (gfx1250) — compile-verified
//
#include <hip/hip_runtime.h>
#include <hip/hip_bf16.h>

typedef __attribute__((ext_vector_type(2))) float v2f;
typedef __attribute__((ext_vector_type(4))) float v4f;
typedef __attribute__((ext_vector_type(8))) float v8f;
typedef __attribute__((ext_vector_type(4))) unsigned int v4u;
typedef __attribute__((ext_vector_type(4))) int v4i;
typedef __attribute__((ext_vector_type(8))) int v8i;

#define TT 16384   // B*S tokens
#define HH 2048    // hidden
#define EE 128     // experts
#define KC 64      // K-chunk staged in LDS
#define LDST (KC + 4)          // padded dword stride: 68 -> conflict-free b64 reads

#define HAVE_TDM __has_builtin(__builtin_amdgcn_tensor_load_to_lds)

#if HAVE_TDM
// ---------------------------------------------------------------------------
// Issue one TDM load of weight[0:128, kc:kc+KC] (f32, row stride HH) into LDS
// at byte offset lds_off, with HW padding of 4 dwords after every 64 dwords
// (reproduces the LDST=68 row stride). 2D tensor -> groups 2/3/4 unused (zero).
// 6-arg builtin form (amdgpu-toolchain clang-23).
// ---------------------------------------------------------------------------
__device__ __forceinline__ void tdm_load_chunk(const float* w, int kc,
                                               unsigned lds_off) {
    unsigned long long ga = (unsigned long long)(const void*)(w + kc);
    v4u g0 = {
        1u,                                    // count=1, is_restore=0, gather=0
        lds_off,                               // lds_addr (bytes)
        (unsigned)ga,                          // global_addr[31:0]
        (unsigned)(ga >> 32) | (2u << 30)      // global_addr[56:32] | type=2
    };
    v8i g1 = {
        (2 << 16) | (1 << 20) | (5 << 22) | (3 << 25),
        //  ^data_size=4B  ^pad_en  ^interval=64dw  ^amount=4dw ; wg_mask=0
        (int)(KC << 16),                       // tensor_dim0[15:0]=64 ; abar=0
        (int)(EE << 16),                       // tensor_dim1[15:0]=128; dim0 hi=0
        (int)(KC << 16),                       // tile_dim0=64 ; dim1 hi=0
        (int)EE,                               // tile_dim1=128 ; tile_dim2=0
        (int)HH,                               // tensor_dim0_stride=2048 (lo)
        0,                                     // stride hi / dim1_stride lo
        0
    };
    v4i gz4 = { 0, 0, 0, 0 };
    v8i gz8 = { 0, 0, 0, 0, 0, 0, 0, 0 };
    __builtin_amdgcn_tensor_load_to_lds(g0, g1, gz4, gz4, gz8, 0);
}
#endif

// ---------------------------------------------------------------------------
// Kernel 1: router logits GEMM, fp32 via V_WMMA_F32_16X16X4_F32.
// Block = 256 threads = 8 waves; wave computes 32 tokens x 128 experts.
// Weight chunks DMA'd into double-buffered LDS by the Tensor Data Mover.
// ---------------------------------------------------------------------------
__global__ __launch_bounds__(256) void router_gemm(const float* __restrict__ x,
                                                   const float* __restrict__ w,
                                                   float* __restrict__ logits) {
    __shared__ float wlds[2][EE * LDST];

    const int tid  = threadIdx.x;
    const int wave = tid >> 5;
    const int lane = tid & 31;
    const int l15  = lane & 15;
    const int hi   = lane >> 4;            // 0: K={0,1}, 1: K={2,3}
    const int tokBase = blockIdx.x * 256 + wave * 32;
    const int nchunks = HH / KC;           // 32

    v8f acc[2][8];
    v8f zero = {0.f, 0.f, 0.f, 0.f, 0.f, 0.f, 0.f, 0.f};
#pragma unroll
    for (int m = 0; m < 2; ++m)
#pragma unroll
        for (int n = 0; n < 8; ++n) acc[m][n] = zero;

    // A-operand base pointers (16x4 f32 layout: lane0-15 M=lane K0/K1, lane16-31 K2/K3)
    const float* xa0 = x + (size_t)(tokBase + l15) * HH + 2 * hi;
    const float* xa1 = xa0 + (size_t)16 * HH;

    const unsigned ldsOff0 = (unsigned)(size_t)(void*)&wlds[0][0];
    const unsigned ldsOff1 = (unsigned)(size_t)(void*)&wlds[1][0];

    // One chunk: prefetch next into 'nextOff', wait for current, compute on 'bl'
    auto step = [&](int c, const float* bl, unsigned nextOff) {
        const int kc = c * KC;
#if HAVE_TDM
        if (wave == 0) {
            if (c + 1 < nchunks) {
                tdm_load_chunk(w, kc + KC, nextOff);   // background DMA
                __builtin_amdgcn_s_wait_tensorcnt(1);  // chunk c complete
            } else {
                __builtin_amdgcn_s_wait_tensorcnt(0);
            }
        }
#else
        // fallback: cooperative synchronous staging into bl's buffer
        float* blw = const_cast<float*>(bl);
#pragma unroll
        for (int i = 0; i < 8; ++i) {
            int f4  = tid + 256 * i;
            int row = f4 >> 4;
            int c4  = (f4 & 15) << 2;
            v4f v = *(const v4f*)(w + (size_t)row * HH + kc + c4);
            *(v4f*)(blw + row * LDST + c4) = v;
        }
#endif
        __syncthreads();   // chunk c visible to all waves

#pragma unroll 4
        for (int kk = 0; kk < KC; kk += 4) {
            v2f a0 = *(const v2f*)(xa0 + kc + kk);   // global_load_b64
            v2f a1 = *(const v2f*)(xa1 + kc + kk);
            // issue all 8 B fragments first (distinct regs -> deep DS clause)
            v2f b[8];
#pragma unroll
            for (int n = 0; n < 8; ++n)
                b[n] = *(const v2f*)(bl + (16 * n + l15) * LDST + kk + 2 * hi);
#pragma unroll
            for (int n = 0; n < 8; ++n) {
                acc[0][n] = __builtin_amdgcn_wmma_f32_16x16x4_f32(
                    false, a0, false, b[n], (short)0, acc[0][n], false, false);
                acc[1][n] = __builtin_amdgcn_wmma_f32_16x16x4_f32(
                    false, a1, false, b[n], (short)0, acc[1][n], false, false);
            }
        }
        __syncthreads();   // all waves done reading bl before chunk c+2 lands in it
    };

#if HAVE_TDM
    if (wave == 0) tdm_load_chunk(w, 0, ldsOff0);
#endif

    // chunk loop, manually unrolled x2 so each body has a constant LDS base
    for (int c = 0; c < nchunks; c += 2) {
        step(c,     &wlds[0][0], ldsOff1);
        step(c + 1, &wlds[1][0], ldsOff0);
    }

    // --- store logits: C/D layout: lane holds N=16n+l15, rows M = r + 8*hi
#pragma unroll
    for (int m = 0; m < 2; ++m) {
        int tok0 = tokBase + m * 16;
#pragma unroll
        for (int r = 0; r < 8; ++r) {
            float* dst = logits + (size_t)(tok0 + r + 8 * hi) * EE + l15;
#pragma unroll
            for (int n = 0; n < 8; ++n) dst[16 * n] = acc[m][n][r];
        }
    }
}

// ---------------------------------------------------------------------------
// Kernel 2: softmax + top-8 + renormalize. One wave32 per token.
// top value_i (normalized) = exp(l_i - max) / sum_top8 exp(l_j - max)
// (global softmax denominator cancels in the renormalization)
// ---------------------------------------------------------------------------
__global__ __launch_bounds__(256) void router_topk(const float* __restrict__ logits,
                                                   float* __restrict__ topv,
                                                   int* __restrict__ topi) {
    const int lane = threadIdx.x & 31;
    const int wave = threadIdx.x >> 5;
    const int tok  = blockIdx.x * 8 + wave;

    v4f l = *(const v4f*)(logits + (size_t)tok * EE + lane * 4);

    // wave-wide max
    float m = fmaxf(fmaxf(l.x, l.y), fmaxf(l.z, l.w));
#pragma unroll
    for (int off = 16; off; off >>= 1) m = fmaxf(m, __shfl_xor(m, off, 32));

    float p0 = __expf(l.x - m);
    float p1 = __expf(l.y - m);
    float p2 = __expf(l.z - m);
    float p3 = __expf(l.w - m);

    const int idx0 = lane * 4;
    float myv = 0.f;
    int   myi = 0;
    float tsum = 0.f;

#pragma unroll
    for (int it = 0; it < 8; ++it) {
        // local argmax over 4 (strict > keeps the lower index on ties)
        float bv = p0; int bi = idx0;
        if (p1 > bv) { bv = p1; bi = idx0 + 1; }
        if (p2 > bv) { bv = p2; bi = idx0 + 2; }
        if (p3 > bv) { bv = p3; bi = idx0 + 3; }
        // wave argmax, tie -> lower index (matches jax top_k stability)
#pragma unroll
        for (int off = 16; off; off >>= 1) {
            float ov = __shfl_xor(bv, off, 32);
            int   oi = __shfl_xor(bi, off, 32);
            if (ov > bv || (ov == bv && oi < bi)) { bv = ov; bi = oi; }
        }
        if (lane == it) { myv = bv; myi = bi; }
        tsum += bv;
        // knock out the winner in its owning lane
        if (bi == idx0)     p0 = -1.f;
        if (bi == idx0 + 1) p1 = -1.f;
        if (bi == idx0 + 2) p2 = -1.f;
        if (bi == idx0 + 3) p3 = -1.f;
    }

    if (lane < 8) {
        topv[(size_t)tok * 8 + lane] = myv / tsum;
        topi[(size_t)tok * 8 + lane] = myi;
    }
}

// ---------------------------------------------------------------------------
extern "C" void kernel_launch(void* const* d_in, const int* in_sizes, int n_in,
                              void* d_out, int out_size, void* d_ws, size_t ws_size,
                              hipStream_t stream) {
    const float* x = (const float*)d_in[0];   // [B,S,H] f32
    const float* w = (const float*)d_in[1];   // [E,H]  f32
    // d_in[2] is top_k scalar (==8), fixed at compile time here

    float* logits = (float*)d_out;                         // [T,E]
    float* topv   = logits + (size_t)TT * EE;              // [T,8]
    int*   topi   = (int*)(topv + (size_t)TT * 8);         // [T,8]

    router_gemm<<<TT / 256, 256, 0, stream>>>(x, w, logits);
    router_topk<<<TT / 8, 256, 0, stream>>>(logits, topv, topi);
}